// ClustGeoEdgeEncoder_61555471286689
// MI455X (gfx1250) — compile-verified
//
#include <hip/hip_runtime.h>
#include <hip/hip_bf16.h>

typedef __attribute__((ext_vector_type(2))) float v2f;
typedef __attribute__((ext_vector_type(8))) float v8f;

#define NPTS   262144
#define NCL    2048
#define PPC    128    // points per cluster
#define NEDGE  8192

// Map float bits to a monotonic unsigned key (ascending uint order == ascending float order).
__device__ __forceinline__ unsigned int fkey(float f) {
    unsigned int u = __float_as_uint(f);
    return u ^ (((unsigned int)((int)u >> 31)) | 0x80000000u);
}

__launch_bounds__(256, 2)
__global__ void clust_geo_edge_kernel(const float* __restrict__ data,
                                      const int*   __restrict__ clusts,
                                      const int*   __restrict__ edge_index,
                                      float*       __restrict__ out)
{
    __shared__ float4             s_x1[PPC];   // (x, y, z, 0)
    __shared__ float4             s_x2[PPC];   // (x, y, z, |p|^2)
    __shared__ float              s_n1[PPC];   // |p|^2 for cluster 1
    __shared__ unsigned long long s_wbest[8];

    const int e = blockIdx.x;
    const int t = threadIdx.x;

    const int c1 = edge_index[e];          // edge_index[0][e]
    const int c2 = edge_index[NEDGE + e];  // edge_index[1][e]

    // ---- stage both clusters into LDS ----
    if (t < PPC) {
        const int pi = clusts[c1 * PPC + t];
        const float x = data[pi * 5 + 0];
        const float y = data[pi * 5 + 1];
        const float z = data[pi * 5 + 2];
        s_x1[t] = make_float4(x, y, z, 0.0f);
        s_n1[t] = x * x + y * y + z * z;
    } else {
        const int tt = t - PPC;
        const int pi = clusts[c2 * PPC + tt];
        const float x = data[pi * 5 + 0];
        const float y = data[pi * 5 + 1];
        const float z = data[pi * 5 + 2];
        s_x2[tt] = make_float4(x, y, z, x * x + y * y + z * z);
    }
    __syncthreads();

    const int wave = t >> 5;     // 0..7 -> tile row ti
    const int lane = t & 31;
    const int lo   = lane & 15;
    const int hi   = lane >> 4;  // 0 or 1

    // A fragment for row i = ti*16+lo: (-2x, -2y, -2z, 1)
    //   VGPR0 = {K0 | K2} = {-2x | -2z},  VGPR1 = {K1 | K3} = {-2y | 1}
    const int ti = wave;
    const float4 arow = s_x1[ti * 16 + lo];
    v2f a;
    a.x = hi ? (-2.0f * arow.z) : (-2.0f * arow.x);
    a.y = hi ? 1.0f             : (-2.0f * arow.y);

    // C init: C[r] = |x1_i|^2 for i = ti*16 + r + 8*hi  (8 one-time LDS loads)
    v8f cinit;
#pragma unroll
    for (int r = 0; r < 8; ++r) cinit[r] = s_n1[ti * 16 + r + 8 * hi];

    unsigned long long best = 0xFFFFFFFFFFFFFFFFull;

    for (int tj = 0; tj < 8; ++tj) {
        // B fragment for col j = tj*16+lo: (x, y, z, |x2_j|^2)
        //   VGPR0 = {x | z},  VGPR1 = {y | n2}
        const float4 b4 = s_x2[tj * 16 + lo];
        v2f b;
        b.x = hi ? b4.z : b4.x;
        b.y = hi ? b4.w : b4.y;

        // D_ij = |x1_i|^2 - 2 x1_i . x2_j + |x2_j|^2  == squared distance, directly
        v8f d = __builtin_amdgcn_wmma_f32_16x16x4_f32(
                    /*neg_a=*/false, a, /*neg_b=*/false, b,
                    /*c_mod=*/(short)0, cinit, /*reuse_a=*/false, /*reuse_b=*/false);

#pragma unroll
        for (int r = 0; r < 8; ++r) {
            const unsigned int i   = (unsigned)(ti * 16 + r + 8 * hi);
            const unsigned int j   = (unsigned)(tj * 16 + lo);
            const unsigned int idx = i * PPC + j;   // row-major flat index
            const unsigned long long key =
                ((unsigned long long)fkey(d[r]) << 32) | (unsigned long long)idx;
            best = (key < best) ? key : best;       // branchless u64 argmin w/ tie-break
        }
    }

    // ---- wave32 branchless min reduction ----
#pragma unroll
    for (int off = 16; off > 0; off >>= 1) {
        const unsigned long long o = __shfl_xor(best, off, 32);
        best = (o < best) ? o : best;
    }
    if (lane == 0) s_wbest[wave] = best;
    __syncthreads();

    // ---- cross-wave reduction + output (19 floats) ----
    if (t == 0) {
        unsigned long long bk = s_wbest[0];
#pragma unroll
        for (int w = 1; w < 8; ++w) bk = (s_wbest[w] < bk) ? s_wbest[w] : bk;

        const unsigned int bi = (unsigned int)(bk & 0xFFFFFFFFu);
        const int i1 = (int)(bi / PPC), i2 = (int)(bi % PPC);
        const float4 v1 = s_x1[i1];
        const float4 v2 = s_x2[i2];
        const float dx = v1.x - v2.x, dy = v1.y - v2.y, dz = v1.z - v2.z;
        const float lend = sqrtf(dx * dx + dy * dy + dz * dz);
        const float inv  = (lend > 0.0f) ? (1.0f / lend) : 1.0f;
        const float nx = dx * inv, ny = dy * inv, nz = dz * inv;

        float* o = out + (size_t)e * 19;
        o[0]  = v1.x; o[1]  = v1.y; o[2]  = v1.z;
        o[3]  = v2.x; o[4]  = v2.y; o[5]  = v2.z;
        o[6]  = nx;   o[7]  = ny;   o[8]  = nz;
        o[9]  = lend;
        o[10] = nx * nx; o[11] = nx * ny; o[12] = nx * nz;
        o[13] = ny * nx; o[14] = ny * ny; o[15] = ny * nz;
        o[16] = nz * nx; o[17] = nz * ny; o[18] = nz * nz;
    }
}

extern "C" void kernel_launch(void* const* d_in, const int* in_sizes, int n_in,
                              void* d_out, int out_size, void* d_ws, size_t ws_size,
                              hipStream_t stream) {
    const float* data       = (const float*)d_in[0];  // [N_PTS, 5] f32
    const int*   clusts     = (const int*)d_in[1];    // [N_CLUSTS, P] i32
    const int*   edge_index = (const int*)d_in[2];    // [2, E] i32
    float*       out        = (float*)d_out;          // [E, 19] f32
    (void)in_sizes; (void)n_in; (void)out_size; (void)d_ws; (void)ws_size;

    clust_geo_edge_kernel<<<NEDGE, 256, 0, stream>>>(data, clusts, edge_index, out);
}